// LogNormalCRPSLoss_34359738617
// MI455X (gfx1250) — compile-verified
//
#include <hip/hip_runtime.h>
#include <math.h>

typedef __attribute__((ext_vector_type(2))) float v2f;
typedef __attribute__((ext_vector_type(8))) float v8f;

#define EPS_F 1e-6f

// One wave (32 lanes, wave32) handles a 16-column tile of n.
// 8 waves / block -> 128 columns per block. N = 32768 -> 256 blocks.
// WMMA f32 16x16x4 with A = ones(16x4) performs the sum-over-samples
// reduction for all 16 columns at once: D[m][n] = sum_k B[k][n] (+C).
// S is a compile-time constant so all loops have exact trip counts
// (S = 100 is divisible by 4 -> no remainder paths).
template <int S>
__global__ __launch_bounds__(256) void crps_kernel(
    const float* __restrict__ mu, const float* __restrict__ sigma,
    const float* __restrict__ target, const float* __restrict__ noise,
    float* __restrict__ block_out, int N)
{
    __shared__ float smem[8 * S * 16];   // per-wave sample tiles [s][c]
    __shared__ float wave_part[8];

    const int tid  = threadIdx.x;
    const int wave = tid >> 5;
    const int lane = tid & 31;
    const int r    = lane >> 4;   // lane-half: selects which K rows this lane feeds
    const int c    = lane & 15;   // column within the 16-wide tile
    const int n    = blockIdx.x * 128 + wave * 16 + c;

    float* tile = &smem[wave * S * 16];

    const float m  = mu[n];
    const float sg = fmaxf(sigma[n], EPS_F);
    const float tg = fmaxf(target[n], EPS_F);

    const v2f onesA = {1.0f, 1.0f};
    v8f acc1 = {};   // column sums of |x_s - t|

    // ---- Phase 1: samples -> LDS, term1 reduction on the matrix pipe ----
    // 4 sample rows per iteration: lane-half r produces rows s0+2r, s0+2r+1.
#pragma unroll 5
    for (int s0 = 0; s0 < S; s0 += 4) {
        const int sa = s0 + 2 * r;
        const float xa = __expf(m + sg * noise[(size_t)sa       * N + n]);
        const float xb = __expf(m + sg * noise[(size_t)(sa + 1) * N + n]);
        tile[sa * 16 + c]       = xa;
        tile[(sa + 1) * 16 + c] = xb;
        v2f b = { fabsf(xa - tg), fabsf(xb - tg) };
        acc1 = __builtin_amdgcn_wmma_f32_16x16x4_f32(
            false, onesA, false, b, (short)0, acc1, false, false);
    }
    __syncthreads();   // tile visible to all lanes of the wave (and block)

    // ---- Phase 2: pairwise term. Per lane: full-j row sums for 2 i's,
    // then WMMA-reduce the 4 row-sums per column per iteration. ----
    v8f acc2 = {};
    for (int i0 = 0; i0 < S; i0 += 4) {
        const int ia = i0 + 2 * r;
        const float xi0 = tile[ia * 16 + c];
        const float xi1 = tile[(ia + 1) * 16 + c];
        float p0 = 0.0f, p1 = 0.0f;
#pragma unroll 10
        for (int j = 0; j < S; ++j) {
            const float xj = tile[j * 16 + c];   // same address for lanes c and c+16
            p0 += fabsf(xi0 - xj);
            p1 += fabsf(xi1 - xj);
        }
        v2f b = { p0, p1 };
        acc2 = __builtin_amdgcn_wmma_f32_16x16x4_f32(
            false, onesA, false, b, (short)0, acc2, false, false);
    }

    // Every C/D row holds the column sum (A was all-ones); acc[0] is D[M][c].
    constexpr float invS = 1.0f / (float)S;
    const float term1 = acc1[0] * invS;
    const float term2 = 0.5f * acc2[0] * invS * invS;
    float v = term1 - term2;   // duplicated across the two lane-halves

    // Wave reduction: 32-lane sum counts each column twice -> halve.
    for (int off = 16; off > 0; off >>= 1)
        v += __shfl_xor(v, off, 32);
    if (lane == 0) wave_part[wave] = v * 0.5f;
    __syncthreads();

    if (tid == 0) {
        float s = 0.0f;
        for (int w = 0; w < 8; ++w) s += wave_part[w];
        block_out[blockIdx.x] = s;
    }
}

__global__ __launch_bounds__(256) void final_reduce(
    const float* __restrict__ block_out, float* __restrict__ out,
    int nblocks, float invN)
{
    __shared__ float sm[256];
    const int tid = threadIdx.x;
    float v = 0.0f;
    for (int i = tid; i < nblocks; i += 256) v += block_out[i];
    sm[tid] = v;
    __syncthreads();
    for (int off = 128; off > 0; off >>= 1) {
        if (tid < off) sm[tid] += sm[tid + off];
        __syncthreads();
    }
    if (tid == 0) out[0] = sm[0] * invN;
}

extern "C" void kernel_launch(void* const* d_in, const int* in_sizes, int n_in,
                              void* d_out, int out_size, void* d_ws, size_t ws_size,
                              hipStream_t stream) {
    const float* mu     = (const float*)d_in[0];
    const float* sigma  = (const float*)d_in[1];
    const float* target = (const float*)d_in[2];
    const float* noise  = (const float*)d_in[3];
    const int N = in_sizes[0];
    const int nblocks = N / 128;            // 256 for N = 32768

    float* block_out = (float*)d_ws;
    // Reference fixes S = 100 (in_sizes[3] / N); compile-time instantiation.
    crps_kernel<100><<<nblocks, 256, 0, stream>>>(mu, sigma, target, noise,
                                                  block_out, N);
    final_reduce<<<1, 256, 0, stream>>>(block_out, (float*)d_out,
                                        nblocks, 1.0f / (float)N);
}